// HistoricalFeatureProjector_22230750724096
// MI455X (gfx1250) — compile-verified
//
#include <hip/hip_runtime.h>
#include <hip/hip_bf16.h>

typedef float v2f __attribute__((ext_vector_type(2)));
typedef float v8f __attribute__((ext_vector_type(8)));

#define VOXEL_SIZE 0.0625f

// ---- tiny affine (4x4 with bottom row 0 0 0 1) helpers, 3x4 row-major ----
__device__ __forceinline__ void affine_inv(const float* M, float* R) {
    float a = M[0], b = M[1], c = M[2];
    float d = M[4], e = M[5], f = M[6];
    float g = M[8], h = M[9], i = M[10];
    float det = a * (e * i - f * h) - b * (d * i - f * g) + c * (d * h - e * g);
    float id = 1.0f / det;
    R[0] = (e * i - f * h) * id;  R[1] = (c * h - b * i) * id;  R[2]  = (b * f - c * e) * id;
    R[4] = (f * g - d * i) * id;  R[5] = (a * i - c * g) * id;  R[6]  = (c * d - a * f) * id;
    R[8] = (d * h - e * g) * id;  R[9] = (b * g - a * h) * id;  R[10] = (a * e - b * d) * id;
    R[3]  = -(R[0] * M[3] + R[1] * M[7] + R[2]  * M[11]);
    R[7]  = -(R[4] * M[3] + R[5] * M[7] + R[6]  * M[11]);
    R[11] = -(R[8] * M[3] + R[9] * M[7] + R[10] * M[11]);
}

__device__ __forceinline__ void affine_mul(const float* A, const float* B, float* R) {
    #pragma unroll
    for (int r = 0; r < 3; ++r) {
        #pragma unroll
        for (int j = 0; j < 4; ++j) {
            float acc = A[r * 4 + 0] * B[0 * 4 + j] + A[r * 4 + 1] * B[1 * 4 + j] +
                        A[r * 4 + 2] * B[2 * 4 + j];
            if (j == 3) acc += A[r * 4 + 3];
            R[r * 4 + j] = acc;
        }
    }
}

// ---- trilinear sampler with grid_sample(align_corners=False, zeros) semantics ----
template <int C, int S>
__device__ __forceinline__ void sample_vol(const float* __restrict__ vol, int b,
                                           float hx, float hy, float hz, float invres,
                                           float* __restrict__ po, int lane) {
    const float Sm1 = (float)(S - 1);
    float gx = fminf(fmaxf((hx * invres) / Sm1 * 2.0f - 1.0f, -1.0f), 1.0f);
    float gy = fminf(fmaxf((hy * invres) / Sm1 * 2.0f - 1.0f, -1.0f), 1.0f);
    float gz = fminf(fmaxf((hz * invres) / Sm1 * 2.0f - 1.0f, -1.0f), 1.0f);

    float ix = ((gx + 1.0f) * (float)S - 1.0f) * 0.5f;
    float iy = ((gy + 1.0f) * (float)S - 1.0f) * 0.5f;
    float iz = ((gz + 1.0f) * (float)S - 1.0f) * 0.5f;

    float fx0 = floorf(ix), fy0 = floorf(iy), fz0 = floorf(iz);
    float fx = ix - fx0, fy = iy - fy0, fz = iz - fz0;
    int x0 = (int)fx0, y0 = (int)fy0, z0 = (int)fz0;   // each in [-1, S-1]

    // per-axis weights with zero-padding masks folded in
    float wx0 = (1.0f - fx) * (x0 >= 0 ? 1.0f : 0.0f);
    float wx1 = fx * (x0 < S - 1 ? 1.0f : 0.0f);
    float wy0 = (1.0f - fy) * (y0 >= 0 ? 1.0f : 0.0f);
    float wy1 = fy * (y0 < S - 1 ? 1.0f : 0.0f);
    float wz0 = (1.0f - fz) * (z0 >= 0 ? 1.0f : 0.0f);
    float wz1 = fz * (z0 < S - 1 ? 1.0f : 0.0f);

    int xa = max(x0, 0), xb = min(x0 + 1, S - 1);
    int ya = max(y0, 0), yb = min(y0 + 1, S - 1);
    int za = max(z0, 0), zb = min(z0 + 1, S - 1);

    int o00 = (za * S + ya) * S;
    int o01 = (za * S + yb) * S;
    int o10 = (zb * S + ya) * S;
    int o11 = (zb * S + yb) * S;

    const size_t S3 = (size_t)S * S * S;
    const float* bp = vol + (size_t)b * C * S3;

    // Fully unroll so up to C/32 * 8 random gathers are in flight before any
    // s_wait_loadcnt -- MLP is the only latency-hiding lever for random access.
    #pragma unroll
    for (int c0 = 0; c0 < C; c0 += 32) {
        int c = c0 + lane;
        if (c < C) {
            const float* pc = bp + (size_t)c * S3;
            float v000 = pc[o00 + xa];
            float v001 = pc[o00 + xb];
            float v010 = pc[o01 + xa];
            float v011 = pc[o01 + xb];
            float v100 = pc[o10 + xa];
            float v101 = pc[o10 + xb];
            float v110 = pc[o11 + xa];
            float v111 = pc[o11 + xb];
            float r = wz0 * (wy0 * (wx0 * v000 + wx1 * v001) +
                             wy1 * (wx0 * v010 + wx1 * v011)) +
                      wz1 * (wy0 * (wx0 * v100 + wx1 * v101) +
                             wy1 * (wx0 * v110 + wx1 * v111));
            // output is 236 MB streamed once: NT store so L2 keeps the grids
            __builtin_nontemporal_store(r, po + c);
        }
    }
}

__global__ __launch_bounds__(256) void hfp_kernel(
    const float* __restrict__ coarse, const float* __restrict__ medium,
    const float* __restrict__ fine, const float* __restrict__ sdf,
    const float* __restrict__ histP, const float* __restrict__ curP,
    const int* __restrict__ vox, float* __restrict__ out, int N) {
    __shared__ float sM[24];        // T_ch_inv for b=0,1 : 3x4 row-major each
    __shared__ int   sB[8][16];     // batch index per point per wave
    __shared__ float sHist[8][96];  // 16 points x 6 candidate coords per wave

    const int lane = threadIdx.x & 31;
    const int wave = threadIdx.x >> 5;
    const int p0 = (blockIdx.x * 8 + wave) * 16;   // 16 points per wave

    // One thread builds both candidate transforms: T_ch_inv = inv(cur @ inv(hist))
    if (threadIdx.x == 0) {
        for (int b = 0; b < 2; ++b) {
            float H[12], Cp[12], iH[12], T[12], Ti[12];
            for (int k = 0; k < 12; ++k) { H[k] = histP[b * 16 + k]; Cp[k] = curP[b * 16 + k]; }
            affine_inv(H, iH);
            affine_mul(Cp, iH, T);
            affine_inv(T, Ti);
            for (int k = 0; k < 12; ++k) sM[b * 12 + k] = Ti[k];
        }
    }

    // ---- WMMA A matrix (16x4 f32): homo coords of 16 points ----
    // ISA layout: lanes 0-15 hold (K=0,K=1) of row M=lane; lanes 16-31 hold (K=2,K=3)
    const int col = lane & 15;
    int pl = p0 + col;
    if (pl >= N) pl = N - 1;
    int4 vi = *(const int4*)(vox + (size_t)pl * 4);   // [b, x, y, z]
    v2f Av;
    if (lane < 16) {
        Av.x = (float)vi.y * VOXEL_SIZE;   // K0 = world x
        Av.y = (float)vi.z * VOXEL_SIZE;   // K1 = world y
        sB[wave][lane] = vi.x;
    } else {
        Av.x = (float)vi.w * VOXEL_SIZE;   // K2 = world z
        Av.y = 1.0f;                       // K3 = 1
    }
    __syncthreads();

    // ---- WMMA B matrix (4x16 f32): column n=3b+i holds row i of M_b ----
    // lanes 0-15: (K=0,K=1) of column N=lane; lanes 16-31: (K=2,K=3) of column N=lane-16
    v2f Bv = {0.0f, 0.0f};
    if (col < 6) {
        int bb = (col >= 3) ? 1 : 0;
        int ii = col - 3 * bb;
        int kb = (lane >> 4) << 1;                 // 0 or 2
        Bv.x = sM[bb * 12 + ii * 4 + kb];
        Bv.y = sM[bb * 12 + ii * 4 + kb + 1];
    }

    v8f acc = {0.f, 0.f, 0.f, 0.f, 0.f, 0.f, 0.f, 0.f};
    // D[p, 3b+i] = hist coord i of point p under transform b
    acc = __builtin_amdgcn_wmma_f32_16x16x4_f32(false, Av, false, Bv,
                                                (short)0, acc, false, false);

    // D layout: VGPR r holds (M=r, N=lane) for lanes 0-15, (M=r+8, N=lane-16) for 16-31
    const int rbase = (lane >> 4) << 3;
    #pragma unroll
    for (int r = 0; r < 8; ++r) {
        if (col < 6) sHist[wave][(rbase + r) * 6 + col] = acc[r];
    }
    __syncthreads();

    // ---- gather phase: lanes over channels, 16 points sequentially ----
    for (int j = 0; j < 16; ++j) {
        if (p0 + j >= N) break;                    // wave-uniform
        int b = sB[wave][j];
        const float* hp = &sHist[wave][j * 6 + 3 * b];
        float hx = hp[0], hy = hp[1], hz = hp[2];
        float* po = out + (size_t)(p0 + j) * 225;
        sample_vol<128, 24>(coarse, b, hx, hy, hz, 4.0f,  po + 0,   lane);
        sample_vol<64,  48>(medium, b, hx, hy, hz, 8.0f,  po + 128, lane);
        sample_vol<32,  96>(fine,   b, hx, hy, hz, 16.0f, po + 192, lane);
        sample_vol<1,   96>(sdf,    b, hx, hy, hz, 16.0f, po + 224, lane);
    }
}

extern "C" void kernel_launch(void* const* d_in, const int* in_sizes, int n_in,
                              void* d_out, int out_size, void* d_ws, size_t ws_size,
                              hipStream_t stream) {
    const float* coarse = (const float*)d_in[0];
    const float* medium = (const float*)d_in[1];
    const float* fine   = (const float*)d_in[2];
    const float* sdf    = (const float*)d_in[3];
    const float* histP  = (const float*)d_in[4];
    const float* curP   = (const float*)d_in[5];
    const int*   vox    = (const int*)d_in[6];
    float* out = (float*)d_out;

    int N = in_sizes[6] / 4;                 // 262144 points
    int blocks = (N + 127) / 128;            // 8 waves/block, 16 points/wave
    hipLaunchKernelGGL(hfp_kernel, dim3(blocks), dim3(256), 0, stream,
                       coarse, medium, fine, sdf, histP, curP, vox, out, N);
}